// Tokenizer_82471962018691
// MI455X (gfx1250) — compile-verified
//
#include <hip/hip_runtime.h>

typedef float v2f __attribute__((ext_vector_type(2)));
typedef float v8f __attribute__((ext_vector_type(8)));

#define NEG_INF (-1.0e9f)

// ---- 16-lane-group reductions (wave32: masks 1,2,4,8 never cross the 16-lane boundary) ----
__device__ __forceinline__ float grp16_max(float v) {
  v = fmaxf(v, __shfl_xor(v, 1, 32));
  v = fmaxf(v, __shfl_xor(v, 2, 32));
  v = fmaxf(v, __shfl_xor(v, 4, 32));
  v = fmaxf(v, __shfl_xor(v, 8, 32));
  return v;
}
__device__ __forceinline__ float grp16_sum(float v) {
  v += __shfl_xor(v, 1, 32);
  v += __shfl_xor(v, 2, 32);
  v += __shfl_xor(v, 4, 32);
  v += __shfl_xor(v, 8, 32);
  return v;
}

// ============================================================================
// Kernel A: 2048 backward lattices. 16 lanes per lattice (lane m owns window
// value w[15-m]); per step: gather phi, group logsumexp -> a_new, merge the
// step (max,sum) into the running logsumexp for mb, shift window via shfl_up.
// log_betas is not an output, so bwd_lengths are irrelevant here.
// Hot-loop choices for MI455X: fast v_exp/v_log intrinsics on the serial
// chain (~1e-5 abs log-space error over 256 steps), b128 loads for the
// contiguous per-lane id/mask streams, global_prefetch_b8 one 16-step tile
// ahead. 1024 waves for latency hiding.
// ============================================================================
__global__ __launch_bounds__(256)
void bwd_mb_kernel(const int* __restrict__ ids, const float* __restrict__ ms,
                   const float* __restrict__ wts, float* __restrict__ mb) {
  const int t = blockIdx.x * 256 + threadIdx.x;   // 0..32767
  const int n = t >> 4;                           // lattice 0..2047
  const int m = t & 15;                           // edge-length index
  const int base = n * 4096 + m * 256;
  const int*   __restrict__ idp = ids + base;
  const float* __restrict__ msp = ms + base;

  float wA   = (m == 0) ? 0.0f : NEG_INF;         // w0 = [-INF,...,-INF, 0]
  float mmax = NEG_INF, msum = 0.0f;              // running logsumexp for mb

  for (int l0 = 0; l0 < 256; l0 += 16) {
    if (l0 + 16 < 256) {                          // CDNA5 global_prefetch_b8
      __builtin_prefetch(idp + l0 + 16, 0, 1);
      __builtin_prefetch(msp + l0 + 16, 0, 1);
    }
#pragma unroll
    for (int c = 0; c < 4; ++c) {
      // contiguous along l per lane -> global_load_b128
      const int4   iv = *(const int4*)  (idp + l0 + c * 4);
      const float4 mv = *(const float4*)(msp + l0 + c * 4);
      const int   idv[4] = {iv.x, iv.y, iv.z, iv.w};
      const float mkv[4] = {mv.x, mv.y, mv.z, mv.w};
#pragma unroll
      for (int j = 0; j < 4; ++j) {
        const float phi = (mkv[j] > 0.0f) ? wts[idv[j]] : NEG_INF;
        const float e   = wA + phi;               // edge_la[m] = w[15-m] + phi
        const float mx  = grp16_max(e);
        const float gs  = grp16_sum(__expf(e - mx));
        const float a_new = mx + __logf(gs);
        // merge step (mx, gs) into running (mmax, msum); masked/unreachable
        // edges sit at <= -1e9 and underflow to exactly 0 on both sides.
        const float nm = fmaxf(mmax, mx);
        msum = msum * __expf(mmax - nm) + gs * __expf(mx - nm);
        mmax = nm;
        // window shift: w'[15-m] = w_old[16-m]; lane m==0 takes a_new
        const float up = __shfl_up(wA, 1, 16);
        wA = (m == 0) ? a_new : up;
      }
    }
  }
  if (m == 0) mb[n] = mmax + __logf(msum);
}

// ============================================================================
// Kernel B: forward DP + expectation-semiring entropy on the 8 forward
// lattices (128 threads, one block). Emits edge_log_alpha into workspace.
// Precise expf/logf here: this directly produces the validated ent/log_alpha
// outputs and costs nothing at this size.
// ============================================================================
__global__ __launch_bounds__(128)
void fwd_ent_kernel(const int* __restrict__ ids, const float* __restrict__ ms,
                    const int* __restrict__ lengths, const float* __restrict__ wts,
                    float* __restrict__ out, float* __restrict__ ela,
                    float* __restrict__ la_ws) {
  const int t = threadIdx.x;                      // 0..127
  const int b = t >> 4;
  const int m = t & 15;
  const int base = b * 4096 + m * 256;
  const int*   __restrict__ idp = ids + base;
  const float* __restrict__ msp = ms + base;
  float*       __restrict__ elp = ela + base;

  float wA = (m == 0) ? 0.0f : NEG_INF;           // alpha window
  float wR = 0.0f;                                // rbar window (init zeros)
  int idx = lengths[b] - 1;
  idx = idx < 0 ? 0 : (idx > 255 ? 255 : idx);
  float lav = 0.0f, rv = 0.0f;

  for (int l = 0; l < 256; ++l) {
    const int   id  = idp[l];
    const float mk  = msp[l];
    const float phi = (mk > 0.0f) ? wts[id] : NEG_INF;
    const float e   = wA + phi;
    elp[l] = e;                                   // edge_log_alpha[b,m,l]
    const float mx = grp16_max(e);
    const float se = expf(e - mx);
    const float gs = grp16_sum(se);
    const float a_new = mx + logf(gs);
    // p = exp(e - a_new) = se/gs ; live gate at -INF/2
    const float contrib = (e > -0.5e9f) ? (se / gs) * (wR + phi) : 0.0f;
    const float r_new = grp16_sum(contrib);
    if (l == idx) { lav = a_new; rv = r_new; }
    const float upA = __shfl_up(wA, 1, 16);
    const float upR = __shfl_up(wR, 1, 16);
    wA = (m == 0) ? a_new : upA;
    wR = (m == 0) ? r_new : upR;
  }
  if (m == 0) {
    out[b]     = lav - rv;                        // entropy = logZ - E[score]
    out[8 + b] = lav;                             // log_alpha
    la_ws[b]   = lav;
  }
}

// ============================================================================
// Kernel C: c = mmask * exp(ELA + 1⊗mb_row - log_alpha). The rank-1 broadcast
// add over each 16x16 (M x L-tile) runs on the matrix unit:
//   D = A(16x4: ones col) x B(4x16: mb row) + C(ELA tile)
// via V_WMMA_F32_16X16X4_F32, then elementwise exp + mask.
// One wave per tile; 8 waves/block, 16 blocks = 128 tiles.
// ============================================================================
__global__ __launch_bounds__(256)
void cond_wmma_kernel(const float* __restrict__ ela, const float* __restrict__ mb,
                      const float* __restrict__ la_ws, const float* __restrict__ mmask,
                      float* __restrict__ cout) {
  const int wave = threadIdx.x >> 5;
  const int lane = threadIdx.x & 31;
  const int tile = blockIdx.x * 8 + wave;         // 0..127
  const int b    = tile >> 4;
  const int l0   = (tile & 15) << 4;
  const int col  = lane & 15;                     // N (= l within tile)
  const int half = lane >> 4;                     // rows 0-7 vs 8-15
  const int rbase = b * 4096 + (half * 8) * 256 + l0 + col;

  // C/D frag: lane(col) holds 8 rows (M) in v0..v7
  v8f cfrag;
#pragma unroll
  for (int i = 0; i < 8; ++i) cfrag[i] = ela[rbase + i * 256];

  // A 16x4: A[r][0] = 1, rest 0  (lanes 0-15: k0,k1 ; lanes 16-31: k2,k3)
  v2f afrag; afrag.x = (half == 0) ? 1.0f : 0.0f; afrag.y = 0.0f;
  // B 4x16: B[0][col] = mb[b, l0+col], rest 0
  v2f bfrag; bfrag.x = (half == 0) ? mb[b * 256 + l0 + col] : 0.0f; bfrag.y = 0.0f;

  v8f d = __builtin_amdgcn_wmma_f32_16x16x4_f32(false, afrag, false, bfrag,
                                                (short)0, cfrag, false, false);

  const float la = la_ws[b];
#pragma unroll
  for (int i = 0; i < 8; ++i) {
    const float mk = mmask[rbase + i * 256];
    cout[rbase + i * 256] = mk * expf(d[i] - la);
  }
}

// ============================================================================
// Inputs (setup_inputs order):
// 0 fwd_ids (8,16,256) i32     1 fwd_ms (8,16,256) f32   2 lengths (8,) i32
// 3 bwd_ids (2048,16,256) i32  4 bwd_ms (2048,16,256) f32 5 bwd_lengths (unused)
// 6 mmask (8,16,256) f32       7 emask (== bwd_ms, unused) 8 weights (32000,1) f32
// Output: ent[8] | log_alpha[8] | c[8*16*256]   (32784 f32)
// ============================================================================
extern "C" void kernel_launch(void* const* d_in, const int* in_sizes, int n_in,
                              void* d_out, int out_size, void* d_ws, size_t ws_size,
                              hipStream_t stream) {
  (void)in_sizes; (void)n_in; (void)out_size; (void)ws_size;
  const int*   fwd_ids = (const int*)  d_in[0];
  const float* fwd_ms  = (const float*)d_in[1];
  const int*   lengths = (const int*)  d_in[2];
  const int*   bwd_ids = (const int*)  d_in[3];
  const float* bwd_ms  = (const float*)d_in[4];
  const float* mmask   = (const float*)d_in[6];
  const float* wts     = (const float*)d_in[8];
  float* out = (float*)d_out;

  float* ws    = (float*)d_ws;
  float* mb    = ws;                   // 2048 floats
  float* ela   = ws + 2048;            // 32768 floats
  float* la_ws = ws + 2048 + 32768;    // 8 floats

  bwd_mb_kernel<<<128, 256, 0, stream>>>(bwd_ids, bwd_ms, wts, mb);
  fwd_ent_kernel<<<1, 128, 0, stream>>>(fwd_ids, fwd_ms, lengths, wts, out, ela, la_ws);
  cond_wmma_kernel<<<16, 256, 0, stream>>>(ela, mb, la_ws, mmask, out + 16);
}